// GAT_3418793968075
// MI455X (gfx1250) — compile-verified
//
#include <hip/hip_runtime.h>
#include <hip/hip_bf16.h>

typedef float v2f __attribute__((ext_vector_type(2)));
typedef float v8f __attribute__((ext_vector_type(8)));

#define HEADS 8

// ---------------------------------------------------------------------------
// fp32 WMMA GEMM: C[M,Nc] = A[M,K] * B[K,Nc].  K % 16 == 0, Nc % 64 == 0.
// Block = 256 threads = 8 waves laid out 2 (M) x 4 (N); block tile 32 x 64.
// Both A (32x16) and B (16x64) tiles staged through LDS with coalesced loads;
// fragments read from LDS with conflict-free pitches (17 / 80 floats).
// ---------------------------------------------------------------------------
__global__ __launch_bounds__(256)
void gemm_wmma_f32(const float* __restrict__ A, const float* __restrict__ B,
                   float* __restrict__ C, int M, int K, int Nc) {
  __shared__ float As[32][17];   // pitch 17: column reads conflict-free
  __shared__ float Bs[16][80];   // pitch 80: lane-halves hit disjoint banks

  const int tid   = threadIdx.x;
  const int wave  = tid >> 5;           // 0..7
  const int lane  = tid & 31;
  const int waveM = wave & 1;           // 0..1
  const int waveN = wave >> 1;          // 0..3
  const int m0  = blockIdx.x << 5;      // 32-row block tile
  const int n0  = blockIdx.y << 6;      // 64-col block tile
  const int m0w = m0 + (waveM << 4);
  const int n0w = n0 + (waveN << 4);
  const int lrow  = lane & 15;
  const int khalf = (lane >> 4) << 1;   // lanes 0-15 -> K 0,1 ; 16-31 -> K 2,3

  // A-tile cooperative load mapping: 512 floats, 2 per thread (contiguous pair)
  const int ar = tid >> 3;              // 0..31 row
  const int ac = (tid & 7) << 1;        // 0,2,..,14
  int ara = m0 + ar; if (ara >= M) ara = M - 1;
  // B-tile cooperative load mapping: 1024 floats, one float4 per thread
  const int br = tid >> 4;              // 0..15 k-row
  const int bc = (tid & 15) << 2;       // 0,4,..,60

  v8f acc = {};
  for (int k0 = 0; k0 < K; k0 += 16) {
    {
      const float* Ap = A + (size_t)ara * K + k0 + ac;
      As[ar][ac + 0] = Ap[0];
      As[ar][ac + 1] = Ap[1];
      const float4 bv = *(const float4*)(B + (size_t)(k0 + br) * Nc + n0 + bc);
      *(float4*)&Bs[br][bc] = bv;
      if (k0 + 16 < K)
        __builtin_prefetch(Ap + 16, 0, 1);   // gfx1250 global_prefetch_b8
    }
    __syncthreads();
#pragma unroll
    for (int kk = 0; kk < 16; kk += 4) {
      v2f a, b;
      a.x = As[(waveM << 4) + lrow][kk + khalf + 0];
      a.y = As[(waveM << 4) + lrow][kk + khalf + 1];
      b.x = Bs[kk + khalf + 0][(waveN << 4) + lrow];
      b.y = Bs[kk + khalf + 1][(waveN << 4) + lrow];
      // (neg_a, A, neg_b, B, c_mod, C, reuse_a, reuse_b)
      acc = __builtin_amdgcn_wmma_f32_16x16x4_f32(false, a, false, b,
                                                  (short)0, acc, false, false);
    }
    __syncthreads();
  }

  const int col   = n0w + lrow;
  const int rbase = m0w + ((lane >> 4) << 3);
  if (m0 + 32 <= M) {                    // block-uniform fast path: no guards
#pragma unroll
    for (int v = 0; v < 8; ++v)
      C[(size_t)(rbase + v) * Nc + col] = acc[v];
  } else {
#pragma unroll
    for (int v = 0; v < 8; ++v) {
      int r = rbase + v;
      if (r < M) C[(size_t)r * Nc + col] = acc[v];
    }
  }
}

// ---------------------------------------------------------------------------
__global__ void fill_f32(float* __restrict__ p, size_t n, float v) {
  size_t i = (size_t)blockIdx.x * blockDim.x + threadIdx.x;
  size_t stride = (size_t)gridDim.x * blockDim.x;
  for (; i < n; i += stride) p[i] = v;
}

// e_src[n,h] = sum_c h[n,h,c]*a_src[h,c];  e_dst likewise.
template<int CH>
__global__ void node_attn(const float* __restrict__ h, const float* __restrict__ a_src,
                          const float* __restrict__ a_dst, float* __restrict__ es,
                          float* __restrict__ ed, int nnodes) {
  int t = blockIdx.x * blockDim.x + threadIdx.x;
  if (t >= nnodes * HEADS) return;
  int n = t >> 3, hh = t & 7;
  const float* hp = h + (size_t)n * (HEADS * CH) + hh * CH;
  const float* as = a_src + hh * CH;
  const float* ad = a_dst + hh * CH;
  float s = 0.f, d = 0.f;
#pragma unroll 8
  for (int c = 0; c < CH; ++c) { float v = hp[c]; s = fmaf(v, as[c], s); d = fmaf(v, ad[c], d); }
  es[t] = s; ed[t] = d;
}

__device__ inline void edge_nodes(const int* __restrict__ ei, int e, int E,
                                  int& s, int& d) {
  if (e < E) { s = ei[e]; d = ei[E + e]; }
  else       { s = e - E; d = s; }       // appended self-loops
}

__device__ inline void atomic_max_f32(float* addr, float val) {
  unsigned int* ai = reinterpret_cast<unsigned int*>(addr);
  unsigned int old = *ai;
  while (__uint_as_float(old) < val) {
    unsigned int assumed = old;
    old = atomicCAS(ai, assumed, __float_as_uint(val));
    if (old == assumed) break;
  }
}

// Pass 1: alpha = leaky_relu(e_src[src]+e_dst[dst]); m[dst] = segmax(alpha)
__global__ void edge_alpha_max(const int* __restrict__ ei, const float* __restrict__ es,
                               const float* __restrict__ ed, float* __restrict__ alpha,
                               float* __restrict__ m, int E, int ET) {
  int e = blockIdx.x * blockDim.x + threadIdx.x;
  if (e >= ET) return;
  int s, d; edge_nodes(ei, e, E, s, d);
#pragma unroll
  for (int h = 0; h < HEADS; ++h) {
    float a = es[(size_t)s * HEADS + h] + ed[(size_t)d * HEADS + h];
    a = a > 0.f ? a : 0.2f * a;
    alpha[(size_t)e * HEADS + h] = a;
    atomic_max_f32(&m[(size_t)d * HEADS + h], a);
  }
}

// Pass 2: e = exp(alpha - m[dst]); z[dst] = segsum(e)
__global__ void edge_exp_sum(const int* __restrict__ ei, float* __restrict__ alpha,
                             const float* __restrict__ m, float* __restrict__ z,
                             int E, int ET) {
  int e = blockIdx.x * blockDim.x + threadIdx.x;
  if (e >= ET) return;
  int s, d; edge_nodes(ei, e, E, s, d);
#pragma unroll
  for (int h = 0; h < HEADS; ++h) {
    float ex = __expf(alpha[(size_t)e * HEADS + h] - m[(size_t)d * HEADS + h]);
    alpha[(size_t)e * HEADS + h] = ex;
    atomicAdd(&z[(size_t)d * HEADS + h], ex);
  }
}

// Pass 3: out[dst] += h[src] * (e/z[dst]) ; MEAN folds the head-average.
template<int CH, bool MEAN>
__global__ void edge_aggregate(const int* __restrict__ ei, const float* __restrict__ ealpha,
                               const float* __restrict__ z, const float* __restrict__ hf,
                               float* __restrict__ out, int E, int ET) {
  const int F = HEADS * CH;
  long long w = (long long)blockIdx.x * blockDim.x + threadIdx.x;
  long long total = (long long)ET * F;
  if (w >= total) return;
  int e  = (int)(w / F);
  int hc = (int)(w - (long long)e * F);
  int h  = hc / CH;
  int s, d; edge_nodes(ei, e, E, s, d);
  float coef = ealpha[(size_t)e * HEADS + h] / z[(size_t)d * HEADS + h];
  float msg  = hf[(size_t)s * F + hc] * coef;
  if (MEAN) {
    atomicAdd(&out[(size_t)d * CH + (hc - h * CH)], msg * (1.0f / HEADS));
  } else {
    atomicAdd(&out[(size_t)d * F + hc], msg);
  }
}

template<int F>
__global__ void bias_relu(float* __restrict__ x, const float* __restrict__ b, size_t n) {
  size_t i = (size_t)blockIdx.x * blockDim.x + threadIdx.x;
  if (i >= n) return;
  float v = x[i] + b[i & (size_t)(F - 1)];
  x[i] = v > 0.f ? v : 0.f;
}

// One wave32 per node: 64 classes -> 2 per lane; shfl reductions (wave32!).
__global__ __launch_bounds__(256)
void finalize_logsoftmax(const float* __restrict__ acc, const float* __restrict__ b2,
                         float* __restrict__ out, int nnodes) {
  int node = blockIdx.x * 8 + (threadIdx.x >> 5);
  int lane = threadIdx.x & 31;
  if (node >= nnodes) return;
  float v0 = acc[(size_t)node * 64 + lane]      + b2[lane];
  float v1 = acc[(size_t)node * 64 + 32 + lane] + b2[32 + lane];
  float mx = fmaxf(v0, v1);
#pragma unroll
  for (int off = 16; off; off >>= 1) mx = fmaxf(mx, __shfl_xor(mx, off, 32));
  float se = __expf(v0 - mx) + __expf(v1 - mx);
#pragma unroll
  for (int off = 16; off; off >>= 1) se += __shfl_xor(se, off, 32);
  float lse = mx + __logf(se);
  out[(size_t)node * 64 + lane]      = v0 - lse;
  out[(size_t)node * 64 + 32 + lane] = v1 - lse;
}

// ---------------------------------------------------------------------------
extern "C" void kernel_launch(void* const* d_in, const int* in_sizes, int n_in,
                              void* d_out, int out_size, void* d_ws, size_t ws_size,
                              hipStream_t stream) {
  (void)n_in; (void)out_size; (void)ws_size;
  const float* x   = (const float*)d_in[0];
  const int*   ei  = (const int*)  d_in[1];
  const float* W1  = (const float*)d_in[2];
  const float* a1s = (const float*)d_in[3];
  const float* a1d = (const float*)d_in[4];
  const float* b1  = (const float*)d_in[5];
  const float* W2  = (const float*)d_in[6];
  const float* a2s = (const float*)d_in[7];
  const float* a2d = (const float*)d_in[8];
  const float* b2  = (const float*)d_in[9];
  float* out = (float*)d_out;

  const int NFEAT = 512, NHID = 256, NCLASS = 64;
  const int N  = in_sizes[0] / NFEAT;     // 50000
  const int E  = in_sizes[1] / 2;         // 800000
  const int ET = E + N;                   // + self loops

  float* f = (float*)d_ws;
  size_t off = 0;
  float* h1     = f + off; off += (size_t)N * NHID;             // [N,8,32]
  float* out1   = f + off; off += (size_t)N * NHID;             // layer-1 agg / layer-2 input
  float* h2     = f + off; off += (size_t)N * (HEADS * NCLASS); // [N,8,64]
  float* ealpha = f + off; off += (size_t)ET * HEADS;           // per-edge alpha/e
  float* es     = f + off; off += (size_t)N * HEADS;
  float* edv    = f + off; off += (size_t)N * HEADS;
  float* mbuf   = f + off; off += (size_t)N * HEADS;
  float* zbuf   = f + off; off += (size_t)N * HEADS;
  float* out2   = f + off; off += (size_t)N * NCLASS;

  const int eb = (ET + 255) / 256;

  // ---------------- Layer 1 ----------------
  dim3 g1((N + 31) / 32, NHID / 64);
  gemm_wmma_f32<<<g1, 256, 0, stream>>>(x, W1, h1, N, NFEAT, NHID);
  node_attn<32><<<(N * HEADS + 255) / 256, 256, 0, stream>>>(h1, a1s, a1d, es, edv, N);
  fill_f32<<<2048, 256, 0, stream>>>(mbuf, (size_t)N * HEADS, -3.0e38f);
  fill_f32<<<2048, 256, 0, stream>>>(zbuf, (size_t)N * HEADS, 0.f);
  fill_f32<<<4096, 256, 0, stream>>>(out1, (size_t)N * NHID, 0.f);
  edge_alpha_max<<<eb, 256, 0, stream>>>(ei, es, edv, ealpha, mbuf, E, ET);
  edge_exp_sum  <<<eb, 256, 0, stream>>>(ei, ealpha, mbuf, zbuf, E, ET);
  {
    long long total = (long long)ET * NHID;
    int blocks = (int)((total + 255) / 256);
    edge_aggregate<32, false><<<blocks, 256, 0, stream>>>(ei, ealpha, zbuf, h1, out1, E, ET);
  }
  bias_relu<256><<<(int)(((size_t)N * NHID + 255) / 256), 256, 0, stream>>>(out1, b1, (size_t)N * NHID);

  // ---------------- Layer 2 ----------------
  dim3 g2((N + 31) / 32, (HEADS * NCLASS) / 64);
  gemm_wmma_f32<<<g2, 256, 0, stream>>>(out1, W2, h2, N, NHID, HEADS * NCLASS);
  node_attn<64><<<(N * HEADS + 255) / 256, 256, 0, stream>>>(h2, a2s, a2d, es, edv, N);
  fill_f32<<<2048, 256, 0, stream>>>(mbuf, (size_t)N * HEADS, -3.0e38f);
  fill_f32<<<2048, 256, 0, stream>>>(zbuf, (size_t)N * HEADS, 0.f);
  fill_f32<<<2048, 256, 0, stream>>>(out2, (size_t)N * NCLASS, 0.f);
  edge_alpha_max<<<eb, 256, 0, stream>>>(ei, es, edv, ealpha, mbuf, E, ET);
  edge_exp_sum  <<<eb, 256, 0, stream>>>(ei, ealpha, mbuf, zbuf, E, ET);
  {
    long long total = (long long)ET * (HEADS * NCLASS);
    int blocks = (int)((total + 255) / 256);
    edge_aggregate<64, true><<<blocks, 256, 0, stream>>>(ei, ealpha, zbuf, h2, out2, E, ET);
  }
  finalize_logsoftmax<<<(N + 7) / 8, 256, 0, stream>>>(out2, b2, out, N);
}